// ConvSet_39152921870536
// MI455X (gfx1250) — compile-verified
//
#include <hip/hip_runtime.h>
#include <math.h>

// ---------------- problem constants ----------------
#define A_CAPS   32
#define B_CAPS   32
#define KK       9          // 3x3 taps
#define NCAP     288        // KK * A_CAPS
#define PSIZE    16         // 4x4 pose, also DIM
#define OH       21
#define L_SITES  441        // 21*21
#define CH       512        // B_CAPS * PSIZE
#define NROWS    14112      // L_SITES * B_CAPS
#define NTILES   882        // NROWS / 16

typedef __attribute__((ext_vector_type(2))) float v2f;
typedef __attribute__((ext_vector_type(8))) float v8f;

// ============================================================
// Kernel P: precompute score matrices G[o][n][16], reordered
// weights Wt[o][n][16], and score bias c[o].
//   u[o][f]    = sum_d Wk[d][f] * M[o][d]
//   G[n,o][i,j]= 0.25 * sum_k W[n,o][j,k] * u[o][i*4+k]
//   c[o]       = 0.25 * sum_d bk[d] * M[o][d]
// ============================================================
__global__ __launch_bounds__(256) void precompute_kernel(
    const float* __restrict__ W, const float* __restrict__ M,
    const float* __restrict__ Wk, const float* __restrict__ bk,
    float* __restrict__ G, float* __restrict__ Wt, float* __restrict__ cvec)
{
    int idx = blockIdx.x * 256 + threadIdx.x;
    if (idx >= NCAP * B_CAPS * PSIZE) return;
    int o   = idx / (NCAP * PSIZE);
    int rem = idx % (NCAP * PSIZE);
    int n   = rem >> 4;
    int e   = rem & 15;
    int i   = e >> 2, j = e & 3;
    const float* w = W + ((size_t)n * B_CAPS + o) * PSIZE;   // W[n][o][4][4]
    float g = 0.f;
#pragma unroll
    for (int k = 0; k < 4; ++k) {
        float u = 0.f;
#pragma unroll
        for (int d = 0; d < PSIZE; ++d)
            u += Wk[d * PSIZE + (i * 4 + k)] * M[o * PSIZE + d];
        g += w[j * 4 + k] * u;
    }
    G[(size_t)idx]  = 0.25f * g;          // [o][n][e] layout, e = i*4+j
    Wt[(size_t)idx] = w[e];               // [o][n][e] layout
    if (n == 0 && e == 0) {
        float c = 0.f;
        for (int d = 0; d < PSIZE; ++d) c += bk[d] * M[o * PSIZE + d];
        cvec[o] = 0.25f * c;
    }
}

// ============================================================
// Kernel A: fused im2col + capsule transform + PMA attention.
// One workgroup (256 threads = 8 waves) per spatial site.
// Never materializes the 260MB po tensor.
// ============================================================
__global__ __launch_bounds__(256) void caps_pma_kernel(
    const float* __restrict__ x,  const float* __restrict__ G,
    const float* __restrict__ Wt, const float* __restrict__ cvec,
    const float* __restrict__ Mv, const float* __restrict__ Wv,
    const float* __restrict__ bv, float* __restrict__ Opre)
{
    __shared__ float pose[NCAP * 17];      // stride 17 pad vs LDS bank conflicts
    __shared__ float sc[B_CAPS * NCAP];    // scores / softmax weights [o][n]

    const int l  = blockIdx.x;             // 0..440
    const int oy = l / OH, ox = l % OH;
    const int tid = threadIdx.x;

    // ---- im2col gather: pose[n][e] = x[a*16+e][oy*2+kh][ox*2+kw], n=kk*32+a
    for (int idx = tid; idx < NCAP * PSIZE; idx += 256) {
        int n = idx >> 4, e = idx & 15;
        int kk = n >> 5, a = n & 31;
        int kh = kk / 3, kw = kk % 3;
        int gy = oy * 2 + kh, gx = ox * 2 + kw;
        pose[n * 17 + e] = x[((a * 16 + e) * 43 + gy) * 43 + gx];
    }
    __syncthreads();

    // ---- attention logits: S[o][n] = <pose[n], G[o][n]> + c[o]  (pre-scaled)
    for (int idx = tid; idx < B_CAPS * NCAP; idx += 256) {
        int o = idx / NCAP, n = idx % NCAP;
        const float* g = G + (size_t)idx * PSIZE;
        const float* p = pose + n * 17;
        float s = cvec[o];
#pragma unroll
        for (int e = 0; e < PSIZE; ++e) s += p[e] * g[e];
        sc[idx] = s;
    }
    __syncthreads();

    // ---- per-o: softmax over n (wave32 shuffle reduce), weighted 4x4
    //      transform accumulation, then Wv projection + seed add.
    const int wave = tid >> 5, lane = tid & 31;
#pragma unroll 1
    for (int o = wave; o < B_CAPS; o += 8) {
        const float* so = sc + o * NCAP;
        float mx = -3.4e38f;
#pragma unroll
        for (int t = 0; t < 9; ++t) mx = fmaxf(mx, so[t * 32 + lane]);
#pragma unroll
        for (int off = 16; off > 0; off >>= 1)
            mx = fmaxf(mx, __shfl_xor(mx, off, 32));
        float ev[9];
        float sum = 0.f;
#pragma unroll
        for (int t = 0; t < 9; ++t) { ev[t] = __expf(so[t * 32 + lane] - mx); sum += ev[t]; }
#pragma unroll
        for (int off = 16; off > 0; off >>= 1)
            sum += __shfl_xor(sum, off, 32);
        const float inv = 1.0f / sum;

        float y[PSIZE];
#pragma unroll
        for (int e = 0; e < PSIZE; ++e) y[e] = 0.f;
#pragma unroll 1
        for (int t = 0; t < 9; ++t) {
            int n = t * 32 + lane;
            float aat = ev[t] * inv;
            const float* w = Wt + ((size_t)o * NCAP + n) * PSIZE;
            const float* p = pose + n * 17;
#pragma unroll
            for (int i = 0; i < 4; ++i)
#pragma unroll
                for (int k = 0; k < 4; ++k) {
                    float acc2 = p[i * 4 + 0] * w[0 * 4 + k] + p[i * 4 + 1] * w[1 * 4 + k]
                               + p[i * 4 + 2] * w[2 * 4 + k] + p[i * 4 + 3] * w[3 * 4 + k];
                    y[i * 4 + k] += aat * acc2;
                }
        }
        // cross-lane sum: every lane ends with the full 16-vector
#pragma unroll
        for (int e = 0; e < PSIZE; ++e)
#pragma unroll
            for (int off = 16; off > 0; off >>= 1)
                y[e] += __shfl_xor(y[e], off, 32);

        if (lane < PSIZE) {
            int d = lane;
            float acc = Mv[o * PSIZE + d] + bv[d];
#pragma unroll
            for (int e = 0; e < PSIZE; ++e) acc += y[e] * Wv[d * PSIZE + e];
            Opre[((size_t)l * B_CAPS + o) * PSIZE + d] = acc;   // (441,512) layout
        }
    }
}

// ============================================================
// BN stats: biased mean/var over the 441 sites, per channel.
// ============================================================
__global__ __launch_bounds__(256) void bn_stats_kernel(
    const float* __restrict__ X, float* __restrict__ mu, float* __restrict__ rs)
{
    int ch = blockIdx.x * 256 + threadIdx.x;
    if (ch >= CH) return;
    float s = 0.f, s2 = 0.f;
    for (int i = 0; i < L_SITES; ++i) {
        float v = X[(size_t)i * CH + ch];
        s += v; s2 += v * v;
    }
    const float im = 1.0f / (float)L_SITES;
    float m = s * im;
    float var = s2 * im - m * m;
    mu[ch] = m;
    rs[ch] = rsqrtf(var + 1e-5f);
}

// ============================================================
// Kernel B: BN0 + residual relu(O @ Wo^T + bo) via
// V_WMMA_F32_16X16X4_F32 (K=16 in four K=4 steps, full fp32).
// One 16x16 row-tile per wave; rows r = s*32 + o, cols = DIM.
// ============================================================
__global__ __launch_bounds__(256) void bn_res_wmma_kernel(
    const float* __restrict__ Opre, const float* __restrict__ mu0,
    const float* __restrict__ rs0,  const float* __restrict__ g0,
    const float* __restrict__ b0,   const float* __restrict__ Wo,
    const float* __restrict__ bo,   float* __restrict__ Z)
{
    __shared__ float tile[8][256];
    const int wave = threadIdx.x >> 5, lane = threadIdx.x & 31;
    const int t = blockIdx.x * 8 + wave;           // tile id
    const bool active = (t < NTILES);
    const int r0 = t * 16;

    if (active) {
        // load 16x16 tile, apply BN0, stage in LDS
#pragma unroll
        for (int q = 0; q < 8; ++q) {
            int e = q * 32 + lane;
            int m = e >> 4, k = e & 15;
            int r = r0 + m;
            int ch = (r & 31) * PSIZE + k;         // channel = o*16 + d
            float v = Opre[(size_t)r * PSIZE + k];
            tile[wave][e] = g0[ch] * (v - mu0[ch]) * rs0[ch] + b0[ch];
        }
    }
    __syncthreads();

    if (active) {
        const int half = lane >> 4, lm = lane & 15;
        v8f acc = {};
#pragma unroll
        for (int kc = 0; kc < 16; kc += 4) {
            int k0 = kc + half * 2;
            v2f a, b;
            a[0] = tile[wave][lm * 16 + k0];       // A[m=lm, k0]
            a[1] = tile[wave][lm * 16 + k0 + 1];
            b[0] = Wo[lm * 16 + k0];               // B[k, n=lm] = Wo[n][k]
            b[1] = Wo[lm * 16 + k0 + 1];
            acc = __builtin_amdgcn_wmma_f32_16x16x4_f32(
                false, a, false, b, (short)0, acc, false, false);
        }
        // D layout: vgpr j -> M = j + 8*half, N = lm
#pragma unroll
        for (int j = 0; j < 8; ++j) {
            int m = j + 8 * half, nn = lm;
            float obn  = tile[wave][m * 16 + nn];
            float proj = acc[j] + bo[nn];
            Z[(size_t)(r0 + m) * PSIZE + nn] = obn + fmaxf(proj, 0.f);
        }
    }
}

// ============================================================
// Kernel C: BN1 + transpose to NCHW (1, 512, 21, 21)
// ============================================================
__global__ __launch_bounds__(256) void bn_out_kernel(
    const float* __restrict__ Z, const float* __restrict__ mu1,
    const float* __restrict__ rs1, const float* __restrict__ g1,
    const float* __restrict__ b1, float* __restrict__ out)
{
    int tid = blockIdx.x * 256 + threadIdx.x;
    if (tid >= CH * L_SITES) return;
    int ch = tid / L_SITES, s = tid % L_SITES;      // out[ch][s], coalesced store
    float v = Z[(size_t)s * CH + ch];
    out[tid] = g1[ch] * (v - mu1[ch]) * rs1[ch] + b1[ch];
}

// ============================================================
extern "C" void kernel_launch(void* const* d_in, const int* in_sizes, int n_in,
                              void* d_out, int out_size, void* d_ws, size_t ws_size,
                              hipStream_t stream) {
    const float* x  = (const float*)d_in[0];
    const float* W  = (const float*)d_in[1];
    const float* M  = (const float*)d_in[2];
    const float* Wk = (const float*)d_in[3];
    const float* bk = (const float*)d_in[4];
    const float* Wv = (const float*)d_in[5];
    const float* bv = (const float*)d_in[6];
    const float* Wo = (const float*)d_in[7];
    const float* bo = (const float*)d_in[8];
    const float* g0 = (const float*)d_in[9];
    const float* b0 = (const float*)d_in[10];
    const float* g1 = (const float*)d_in[11];
    const float* b1 = (const float*)d_in[12];
    float* out = (float*)d_out;

    float* ws   = (float*)d_ws;
    float* G    = ws;                         // 147456
    float* Wt   = ws + 147456;                // 147456
    float* cv   = ws + 294912;                // 32
    float* Opre = ws + 294944;                // 225792
    float* Z    = ws + 520736;                // 225792
    float* mu0  = ws + 746528;                // 512
    float* rs0  = mu0 + CH;
    float* mu1  = rs0 + CH;
    float* rs1  = mu1 + CH;

    precompute_kernel<<<(NCAP * B_CAPS * PSIZE + 255) / 256, 256, 0, stream>>>(
        W, M, Wk, bk, G, Wt, cv);
    caps_pma_kernel<<<L_SITES, 256, 0, stream>>>(x, G, Wt, cv, M, Wv, bv, Opre);
    bn_stats_kernel<<<2, 256, 0, stream>>>(Opre, mu0, rs0);
    bn_res_wmma_kernel<<<(NTILES + 7) / 8, 256, 0, stream>>>(
        Opre, mu0, rs0, g0, b0, Wo, bo, Z);
    bn_stats_kernel<<<2, 256, 0, stream>>>(Z, mu1, rs1);
    bn_out_kernel<<<(CH * L_SITES + 255) / 256, 256, 0, stream>>>(
        Z, mu1, rs1, g1, b1, out);
}